// Net_67963562492527
// MI455X (gfx1250) — compile-verified
//
#include <hip/hip_runtime.h>

// ---- problem constants (match reference) ----
#define NN   100000          // nodes
#define EE   1600000         // edges (before self loops)
#define ETOT (EE + NN)       // edges incl. self loops
#define F    64              // features (= H*C)
#define HH   8               // heads
#define CC   8               // channels per head
#define NEG_SLOPE 0.2f

typedef __attribute__((ext_vector_type(2))) float v2f;
typedef __attribute__((ext_vector_type(8))) float v8f;

// ---------------------------------------------------------------------------
// Dense transform h = X @ W  (X: [NN,64], W: [64,64]) using fp32 WMMA.
// One wave computes one 16x16 tile of h; block = 4 waves covering the 64 cols.
// A 16x4 f32 fragment (ISA 7.12.2): lanes 0-15 -> M=lane, regs hold K=k0,k0+1;
// lanes 16-31 -> M=lane-16, regs hold K=k0+2,k0+3. B mirrors (K rows).
// C/D: reg r, lanes 0-15 -> row r, lanes 16-31 -> row r+8, col = lane&15.
// ---------------------------------------------------------------------------
__global__ void gemm_wmma(const float* __restrict__ X,
                          const float* __restrict__ W,
                          float* __restrict__ Hout) {
  const int wave = threadIdx.x >> 5;
  const int lane = threadIdx.x & 31;
  const int m0 = blockIdx.x * 16;       // NN=100000 is divisible by 16
  const int n0 = wave * 16;             // 4 waves * 16 = 64 cols
  const int hi = lane >> 4;             // lane group (adds +2 to K, +8 to row)
  const int l  = lane & 15;

  v8f c = {};
  for (int k0 = 0; k0 < F; k0 += 4) {
    const int ka = k0 + hi * 2;
    v2f a, b;
    const float* xr = X + (size_t)(m0 + l) * F + ka;
    a.x = xr[0];
    a.y = xr[1];
    b.x = W[(ka + 0) * F + n0 + l];
    b.y = W[(ka + 1) * F + n0 + l];
    c = __builtin_amdgcn_wmma_f32_16x16x4_f32(false, a, false, b,
                                              (short)0, c, false, false);
  }
  const int col = n0 + l;
#pragma unroll
  for (int r = 0; r < 8; ++r) {
    const int row = m0 + r + hi * 8;
    Hout[(size_t)row * F + col] = c[r];
  }
}

// ---------------------------------------------------------------------------
// Per-node, per-head attention logits: alpha_s/alpha_d [NN, H]
// ---------------------------------------------------------------------------
__global__ void alpha_kernel(const float* __restrict__ Hm,
                             const float* __restrict__ a_src,
                             const float* __restrict__ a_dst,
                             float* __restrict__ as_out,
                             float* __restrict__ ad_out) {
  const int t = blockIdx.x * blockDim.x + threadIdx.x;
  if (t >= NN * HH) return;
  const int hh = t & (HH - 1);
  const int node = t >> 3;
  const float* hp = Hm + (size_t)node * F + hh * CC;
  float s = 0.f, d = 0.f;
#pragma unroll
  for (int ci = 0; ci < CC; ++ci) {
    const float v = hp[ci];
    s += v * a_src[hh * CC + ci];
    d += v * a_dst[hh * CC + ci];
  }
  as_out[t] = s;
  ad_out[t] = d;
}

// ---------------------------------------------------------------------------
// util: fill u32 (also used to zero float buffers)
// ---------------------------------------------------------------------------
__global__ void fill_u32(unsigned* __restrict__ p, unsigned v, int n) {
  const int t = blockIdx.x * blockDim.x + threadIdx.x;
  if (t < n) p[t] = v;
}

// ordered-uint encoding of float so unsigned atomicMax == float max
__device__ __forceinline__ unsigned ford(float f) {
  unsigned u = __float_as_uint(f);
  return (u & 0x80000000u) ? ~u : (u | 0x80000000u);
}
__device__ __forceinline__ float funord(unsigned u) {
  return (u & 0x80000000u) ? __uint_as_float(u ^ 0x80000000u)
                           : __uint_as_float(~u);
}
#define ORD_NEG_INF 0x007FFFFFu  // ford(-inf)

__device__ __forceinline__ void edge_sd(const long long* __restrict__ ei,
                                        int ee, int& s, int& d) {
  if (ee < EE) {
    s = (int)ei[ee];
    d = (int)ei[(size_t)EE + ee];
  } else {
    s = d = ee - EE;  // self loop
  }
}

__device__ __forceinline__ float lrelu(float x) {
  return x > 0.f ? x : NEG_SLOPE * x;
}

// ---------------------------------------------------------------------------
// Pass 1: segment max of edge logits per (dst, head)
// ---------------------------------------------------------------------------
__global__ void edge_max(const long long* __restrict__ ei,
                         const float* __restrict__ as_,
                         const float* __restrict__ ad_,
                         unsigned* __restrict__ mbits) {
  const long long t = (long long)blockIdx.x * blockDim.x + threadIdx.x;
  if (t >= (long long)ETOT * HH) return;
  const int ee = (int)(t >> 3);
  const int hh = (int)(t & 7);
  int s, d;
  edge_sd(ei, ee, s, d);
  const float e = lrelu(as_[s * HH + hh] + ad_[d * HH + hh]);
  atomicMax(&mbits[d * HH + hh], ford(e));
}

// ---------------------------------------------------------------------------
// Pass 2: segment sum of exp(e - m) per (dst, head)
// ---------------------------------------------------------------------------
__global__ void edge_sum(const long long* __restrict__ ei,
                         const float* __restrict__ as_,
                         const float* __restrict__ ad_,
                         const unsigned* __restrict__ mbits,
                         float* __restrict__ denom) {
  const long long t = (long long)blockIdx.x * blockDim.x + threadIdx.x;
  if (t >= (long long)ETOT * HH) return;
  const int ee = (int)(t >> 3);
  const int hh = (int)(t & 7);
  int s, d;
  edge_sd(ei, ee, s, d);
  const float e = lrelu(as_[s * HH + hh] + ad_[d * HH + hh]);
  const float m = funord(mbits[d * HH + hh]);
  atomicAdd(&denom[d * HH + hh], __expf(e - m));
}

// ---------------------------------------------------------------------------
// Pass 3: acc[dst] += h[src] * alpha. One thread per (edge, head*chan) so the
// 64 lanes of an edge read h[src*64 + hc] coalesced and atomically add to
// acc[dst*64 + hc] coalesced (L2-resident on MI455X: h is 25.6MB << 192MB).
// ---------------------------------------------------------------------------
__global__ void edge_agg(const long long* __restrict__ ei,
                         const float* __restrict__ as_,
                         const float* __restrict__ ad_,
                         const unsigned* __restrict__ mbits,
                         const float* __restrict__ denom,
                         const float* __restrict__ Hm,
                         float* __restrict__ acc) {
  const long long t = (long long)blockIdx.x * blockDim.x + threadIdx.x;
  if (t >= (long long)ETOT * F) return;
  const int ee = (int)(t >> 6);
  const int hc = (int)(t & 63);
  const int hh = hc >> 3;
  int s, d;
  edge_sd(ei, ee, s, d);
  const float e = lrelu(as_[s * HH + hh] + ad_[d * HH + hh]);
  const float m = funord(mbits[d * HH + hh]);
  const float alpha = __expf(e - m) / (denom[d * HH + hh] + 1e-16f);
  atomicAdd(&acc[(size_t)d * F + hc], Hm[(size_t)s * F + hc] * alpha);
}

// ---------------------------------------------------------------------------
// out = acc + bias, optional ReLU (in-place safe)
// ---------------------------------------------------------------------------
__global__ void bias_act(const float* __restrict__ acc,
                         const float* __restrict__ b,
                         float* __restrict__ out, int relu) {
  const int t = blockIdx.x * blockDim.x + threadIdx.x;
  if (t >= NN * F) return;
  float v = acc[t] + b[t & 63];
  if (relu) v = fmaxf(v, 0.f);
  out[t] = v;
}

// ---------------------------------------------------------------------------
// host side
// ---------------------------------------------------------------------------
static void run_layer(const float* in, const long long* ei, const float* W,
                      const float* a_src, const float* a_dst, const float* b,
                      float* hBuf, float* asBuf, float* adBuf, unsigned* mBuf,
                      float* dBuf, float* accBuf, float* outBuf, int relu,
                      hipStream_t stream) {
  // 1) h = in @ W  (WMMA)
  gemm_wmma<<<NN / 16, 128, 0, stream>>>(in, W, hBuf);
  // 2) node attention logits
  alpha_kernel<<<(NN * HH + 255) / 256, 256, 0, stream>>>(hBuf, a_src, a_dst,
                                                          asBuf, adBuf);
  // 3) init max / denom / acc
  fill_u32<<<(NN * HH + 255) / 256, 256, 0, stream>>>(mBuf, ORD_NEG_INF, NN * HH);
  fill_u32<<<(NN * HH + 255) / 256, 256, 0, stream>>>((unsigned*)dBuf, 0u, NN * HH);
  fill_u32<<<(NN * F + 255) / 256, 256, 0, stream>>>((unsigned*)accBuf, 0u, NN * F);
  // 4) segment max
  {
    long long nt = (long long)ETOT * HH;
    edge_max<<<(unsigned)((nt + 255) / 256), 256, 0, stream>>>(ei, asBuf, adBuf, mBuf);
  }
  // 5) segment sum of exp
  {
    long long nt = (long long)ETOT * HH;
    edge_sum<<<(unsigned)((nt + 255) / 256), 256, 0, stream>>>(ei, asBuf, adBuf,
                                                               mBuf, dBuf);
  }
  // 6) weighted scatter aggregation
  {
    long long nt = (long long)ETOT * F;
    edge_agg<<<(unsigned)((nt + 255) / 256), 256, 0, stream>>>(ei, asBuf, adBuf,
                                                               mBuf, dBuf, hBuf,
                                                               accBuf);
  }
  // 7) bias (+ReLU)
  bias_act<<<(NN * F + 255) / 256, 256, 0, stream>>>(accBuf, b, outBuf, relu);
}

extern "C" void kernel_launch(void* const* d_in, const int* in_sizes, int n_in,
                              void* d_out, int out_size, void* d_ws,
                              size_t ws_size, hipStream_t stream) {
  const float* x = (const float*)d_in[0];
  const long long* ei = (const long long*)d_in[1];
  const float* W1 = (const float*)d_in[2];
  const float* a_src1 = (const float*)d_in[3];
  const float* a_dst1 = (const float*)d_in[4];
  const float* b1 = (const float*)d_in[5];
  const float* W2 = (const float*)d_in[6];
  const float* a_src2 = (const float*)d_in[7];
  const float* a_dst2 = (const float*)d_in[8];
  const float* b2 = (const float*)d_in[9];
  const float* W3 = (const float*)d_in[10];
  const float* a_src3 = (const float*)d_in[11];
  const float* a_dst3 = (const float*)d_in[12];
  const float* b3 = (const float*)d_in[13];

  // workspace layout (bytes)
  char* ws = (char*)d_ws;
  const size_t szNF = (size_t)NN * F * sizeof(float);   // 25.6 MB
  const size_t szNH = (size_t)NN * HH * sizeof(float);  // 3.2 MB
  float*    hBuf   = (float*)(ws);                       // h (per-layer)
  float*    actBuf = (float*)(ws + szNF);                // layer activations
  float*    asBuf  = (float*)(ws + 2 * szNF);
  float*    adBuf  = (float*)(ws + 2 * szNF + szNH);
  unsigned* mBuf   = (unsigned*)(ws + 2 * szNF + 2 * szNH);
  float*    dBuf   = (float*)(ws + 2 * szNF + 3 * szNH);

  float* out = (float*)d_out;

  // Layer 1: in = x, acc/act in actBuf (ReLU)
  run_layer(x, ei, W1, a_src1, a_dst1, b1, hBuf, asBuf, adBuf, mBuf, dBuf,
            actBuf, actBuf, 1, stream);
  // Layer 2: in = actBuf; acc into actBuf is safe (gemm read precedes zero-fill
  // in stream order), ReLU in place.
  run_layer(actBuf, ei, W2, a_src2, a_dst2, b2, hBuf, asBuf, adBuf, mBuf, dBuf,
            actBuf, actBuf, 1, stream);
  // Layer 3: in = actBuf, accumulate directly into d_out, bias only.
  run_layer(actBuf, ei, W3, a_src3, a_dst3, b3, hBuf, asBuf, adBuf, mBuf, dBuf,
            out, out, 0, stream);
}